// RAMMultiStepTransformer_24309514895615
// MI455X (gfx1250) — compile-verified
//
#include <hip/hip_runtime.h>
#include <stdint.h>

#define NB      16
#define BATCH   256
#define IN_BITS 1024
#define N_IN    2048
#define N_ST    1024
#define N_OUT   512
// packed layout per batch item: words [0..63] = in_bits (2048), [64..95] = state (1024)
#define IW  64
#define SW  32

typedef uint32_t u32x4 __attribute__((ext_vector_type(4)));
typedef int32_t  i32x8 __attribute__((ext_vector_type(8)));
typedef int32_t  i32x4 __attribute__((ext_vector_type(4)));

#ifndef __has_builtin
#define __has_builtin(x) 0
#endif

#if __has_builtin(__builtin_amdgcn_tensor_load_to_lds) && \
    __has_builtin(__builtin_amdgcn_s_wait_tensorcnt)
#define USE_TDM 1
#else
#define USE_TDM 0
#endif

#if USE_TDM
// 1-row tile DMA: copy n_dwords dwords from global to LDS via the Tensor Data Mover.
// Descriptor fields per CDNA5 ISA sec 8.3/8.4.
__device__ __forceinline__ void tdm_load_1d(uint32_t lds_off_bytes,
                                            const void* gptr, uint32_t n_dwords) {
  uint64_t ga = (uint64_t)(uintptr_t)gptr;
  u32x4 g0;
  g0.x = 1u;                                        // count=1 (valid), user mode
  g0.y = lds_off_bytes;                             // lds_addr
  g0.z = (uint32_t)(ga & 0xFFFFFFFFu);              // global_addr[31:0]
  g0.w = (uint32_t)((ga >> 32) & 0x01FFFFFFu)       // global_addr[56:32]
       | 0x80000000u;                               // type=2 ("image") in bits 127:126
  i32x8 g1;
  g1[0] = (int)(2u << 16);                          // data_size=2 (4 bytes); wg_mask=0
  g1[1] = (int)((n_dwords & 0xFFFFu) << 16);        // tensor_dim0[15:0] @bits 63:48
  g1[2] = (int)(((n_dwords >> 16) & 0xFFFFu))       // tensor_dim0[31:16]
        | (1 << 16);                                // tensor_dim1 = 1
  g1[3] = (int)((n_dwords & 0xFFFFu) << 16);        // tile_dim0 @bits 127:112
  g1[4] = 1;                                        // tile_dim1 = 1, tile_dim2 = 0
  g1[5] = (int)n_dwords;                            // tensor_dim0_stride[31:0]
  g1[6] = 0;                                        // stride hi / dim1 stride lo
  g1[7] = 0;
  i32x4 z4 = {0, 0, 0, 0};
  i32x8 z8 = {0, 0, 0, 0, 0, 0, 0, 0};
  // 6-arg form (clang-23 / therock headers): (g0, g1, g2, g3, g4, cpol)
  __builtin_amdgcn_tensor_load_to_lds(g0, g1, z4, z4, z8, 0);
}
#endif

// Stage the packed bit vectors for batch item b into LDS:
// lds[0..63] <- I_b (in_bits), lds[64..95] <- S_b (state). Includes the barrier.
__device__ __forceinline__ void stage_bits(uint32_t* lds,
                                           const uint32_t* I_b,
                                           const uint32_t* S_b) {
#if USE_TDM
  if (threadIdx.x < 32) {                 // one wave issues; EXEC ignored by TDM
    tdm_load_1d((uint32_t)(uintptr_t)lds, I_b, IW);
    tdm_load_1d((uint32_t)(uintptr_t)(lds + IW), S_b, SW);
    __builtin_amdgcn_s_wait_tensorcnt(0);
  }
  __syncthreads();
#else
  int t = threadIdx.x;
  if (t < IW)            lds[t] = I_b[t];
  else if (t < IW + SW)  lds[t] = S_b[t - IW];
  __syncthreads();
#endif
}

// Gather 16 bits from the packed LDS bit vector, form the 16-bit RAM address.
__device__ __forceinline__ uint32_t gather_addr(const uint32_t* lds_bits,
                                                const int* __restrict__ crow) {
  uint32_t addr = 0;
#pragma unroll
  for (int j = 0; j < NB; ++j) {
    int idx = crow[j];
    addr |= ((lds_bits[idx >> 5] >> (idx & 31)) & 1u) << j;
  }
  return addr;
}

// ---------------- Kernel A: input layer ----------------
// grid = BATCH*8 blocks of 256; block (b, chunk) handles neurons [chunk*256, +256).
__global__ void k_input(const int* __restrict__ x,
                        const int* __restrict__ conn_in,
                        const float* __restrict__ mem_in,
                        uint32_t* __restrict__ I,
                        uint32_t* __restrict__ S0) {
  __shared__ uint32_t xb[IN_BITS / 32];   // 32 words = 1024 packed input bits
  const int b     = blockIdx.x >> 3;
  const int chunk = blockIdx.x & 7;
  const int t = threadIdx.x, lane = t & 31, wave = t >> 5;

  // pack x[b] into 32 dwords via wave32 ballot (bit 'lane' of word w = x[w*32+lane])
  const int* xrow = x + (size_t)b * IN_BITS;
#pragma unroll
  for (int k = 0; k < 4; ++k) {
    int w = wave * 4 + k;
    uint32_t m = (uint32_t)__ballot(xrow[w * 32 + lane] != 0);
    if (lane == 0) xb[w] = m;
  }
  __syncthreads();

  const int n = chunk * 256 + t;
  const int* crow = conn_in + (size_t)n * NB;
  uint32_t addr = gather_addr(xb, crow);
  float v = __builtin_nontemporal_load(&mem_in[((size_t)n << 16) + addr]);
  uint32_t m = (uint32_t)__ballot(v > 0.5f);
  if (lane == 0) I[b * IW + chunk * 8 + wave] = m;

  // zero the initial state buffer (deterministic each launch)
  if (chunk == 0 && t < SW) S0[b * SW + t] = 0;
}

// ---------------- Kernel B: one state iteration ----------------
// grid = BATCH*4 blocks of 256; block (b, chunk) handles state neurons [chunk*256, +256).
__global__ void k_state(const int* __restrict__ conn_state,
                        const float* __restrict__ mem_state,
                        const uint32_t* __restrict__ I,
                        const uint32_t* __restrict__ Scur,
                        uint32_t* __restrict__ Snext) {
  __shared__ uint32_t bits[IW + SW];
  const int b     = blockIdx.x >> 2;
  const int chunk = blockIdx.x & 3;
  stage_bits(bits, I + (size_t)b * IW, Scur + (size_t)b * SW);

  const int t = threadIdx.x, lane = t & 31, wave = t >> 5;
  const int n = chunk * 256 + t;
  const int* crow = conn_state + (size_t)n * NB;
  uint32_t addr = gather_addr(bits, crow);
  float v = __builtin_nontemporal_load(&mem_state[((size_t)n << 16) + addr]);
  uint32_t m = (uint32_t)__ballot(v > 0.5f);
  if (lane == 0) Snext[b * SW + chunk * 8 + wave] = m;
}

// ---------------- Kernel C: output layer (final iteration only) ----------------
// grid = BATCH*2 blocks of 256; block (b, chunk) handles out neurons [chunk*256, +256).
__global__ void k_out(const int* __restrict__ conn_out,
                      const float* __restrict__ mem_out,
                      const uint32_t* __restrict__ I,
                      const uint32_t* __restrict__ Sfin,
                      float* __restrict__ out) {
  __shared__ uint32_t bits[IW + SW];
  const int b     = blockIdx.x >> 1;
  const int chunk = blockIdx.x & 1;
  stage_bits(bits, I + (size_t)b * IW, Sfin + (size_t)b * SW);

  const int t = threadIdx.x;
  const int n = chunk * 256 + t;
  const int* crow = conn_out + (size_t)n * NB;
  uint32_t addr = gather_addr(bits, crow);
  float v = __builtin_nontemporal_load(&mem_out[((size_t)n << 16) + addr]);
  out[(size_t)b * N_OUT + n] = v;
}

extern "C" void kernel_launch(void* const* d_in, const int* in_sizes, int n_in,
                              void* d_out, int out_size, void* d_ws, size_t ws_size,
                              hipStream_t stream) {
  const int*   x          = (const int*)d_in[0];
  const int*   conn_in    = (const int*)d_in[1];
  const int*   conn_state = (const int*)d_in[2];
  const int*   conn_out   = (const int*)d_in[3];
  const float* mem_in     = (const float*)d_in[4];
  const float* mem_state  = (const float*)d_in[5];
  const float* mem_out    = (const float*)d_in[6];
  float* out = (float*)d_out;

  // workspace: I (64 KB) | S0 (32 KB) | S1 (32 KB)
  uint32_t* I  = (uint32_t*)d_ws;
  uint32_t* S0 = I + (size_t)BATCH * IW;
  uint32_t* S1 = S0 + (size_t)BATCH * SW;

  k_input<<<BATCH * 8, 256, 0, stream>>>(x, conn_in, mem_in, I, S0);

  uint32_t* cur = S0;
  uint32_t* nxt = S1;
  for (int it = 0; it < 4; ++it) {
    k_state<<<BATCH * 4, 256, 0, stream>>>(conn_state, mem_state, I, cur, nxt);
    uint32_t* tmp = cur; cur = nxt; nxt = tmp;
  }
  // reference overwrites out_vals each iteration -> only final lookup matters
  k_out<<<BATCH * 2, 256, 0, stream>>>(conn_out, mem_out, I, cur, out);
}